// FCOS_80307298501488
// MI455X (gfx1250) — compile-verified
//
#include <hip/hip_runtime.h>
#include <stdint.h>

typedef __attribute__((ext_vector_type(2))) float    v2f;
typedef __attribute__((ext_vector_type(8))) float    v8f;
typedef __attribute__((ext_vector_type(4))) unsigned u32x4;
typedef __attribute__((ext_vector_type(8))) int      i32x8;
typedef __attribute__((ext_vector_type(4))) int      i32x4;

#define K_PRE    2000
#define CAND_CAP 4096
#define NPAD     2048
#define SWORDS   64          // 2048 bits per suppression row
#define IOU_THR  0.5f
#define C_MAX    20          // classes (LDS tile sized for this)
#define TILE_R   256         // anchors per block in score pass

struct SelState {
    unsigned prefix;   // accumulated high bits of K-th largest score
    int      kRem;     // remaining rank within matching set
    unsigned cnt;      // candidate counter
    int      allBg;    // all(deltas == -1) flag
};

// ---------------------------------------------------------------- init
__global__ void initKernel(unsigned* hist, SelState* st, unsigned* S, int sWords) {
    int t = blockIdx.x * blockDim.x + threadIdx.x;
    int stride = gridDim.x * blockDim.x;
    if (t == 0) { st->prefix = 0u; st->kRem = K_PRE; st->cnt = 0u; st->allBg = 1; }
    for (int i = t; i < 4 * 256; i += stride) hist[i] = 0u;
    for (int i = t; i < sWords;  i += stride) S[i] = 0u;
}

// ------------------------------------------------- fused score / argmax / all-bg
// Logits tile staged HBM -> LDS via the Tensor Data Mover (TENSORcnt), then each
// thread scans its row from LDS as float4.
__global__ void scoreKernel(const float* __restrict__ logits,
                            const float* __restrict__ deltas,
                            int N, int C,
                            unsigned* __restrict__ scoreBits,
                            int* __restrict__ classIds,
                            SelState* st) {
    __shared__ float tile[TILE_R * C_MAX];   // 20 KB
    int tid = threadIdx.x;
    int blockStart = blockIdx.x * TILE_R;
    int gid = blockStart + tid;

    bool useTDM = (C == C_MAX) && (blockStart + TILE_R <= N);
#if __has_builtin(__builtin_amdgcn_tensor_load_to_lds)
    if (useTDM) {
        if (tid < 32) {                         // one wave drives the DMA
            unsigned elems   = (unsigned)TILE_R * (unsigned)C;          // 5120
            unsigned long long gaddr =
                (unsigned long long)(uintptr_t)(logits + (size_t)blockStart * C);
            unsigned ldsOff = (unsigned)(uintptr_t)&tile[0];            // LDS byte addr

            // ---- D# group 0 (128b): count=1 | lds_addr | global_addr | type=2
            u32x4 g0;
            g0.x = 1u;                                                  // count=1
            g0.y = ldsOff;                                              // lds_addr
            g0.z = (unsigned)(gaddr & 0xFFFFFFFFull);                   // addr[31:0]
            g0.w = (unsigned)((gaddr >> 32) & 0x1FFFFFFull) | (2u << 30); // addr[56:32]|type=2

            // ---- D# group 1 (256b): data_size=4B, 1-D tile of `elems` elements
            i32x8 g1;
            g1[0] = (int)(2u << 16);                       // data_size=2 (4B), mask=0
            g1[1] = (int)((elems & 0xFFFFu) << 16);        // tensor_dim0[15:0]
            g1[2] = (int)((elems >> 16) & 0xFFFFu);        // tensor_dim0[31:16], dim1=0
            g1[3] = (int)((elems & 0xFFFFu) << 16);        // tile_dim0
            g1[4] = 0;                                     // tile_dim1/2 = 0 (unused)
            g1[5] = 0; g1[6] = 0; g1[7] = 0;               // strides = 0 (1-D)

            i32x4 z4 = {0, 0, 0, 0};
#if __clang_major__ >= 23
            i32x8 z8 = {0, 0, 0, 0, 0, 0, 0, 0};
            __builtin_amdgcn_tensor_load_to_lds(g0, g1, z4, z4, z8, 0);
#else
            __builtin_amdgcn_tensor_load_to_lds(g0, g1, z4, z4, 0);
#endif
            __builtin_amdgcn_s_wait_tensorcnt(0);
        }
        __syncthreads();
    }
#else
    useTDM = false;
#endif

    if (gid >= N) return;

    float m = -3.402823466e38f;
    int am = 0;
    if (useTDM) {
        const float4* row = (const float4*)&tile[tid * C_MAX];  // 80B rows, 16B aligned
        for (int q = 0; q < C_MAX / 4; ++q) {
            float4 v = row[q];
            if (v.x > m) { m = v.x; am = q * 4 + 0; }   // first-max == jnp.argmax
            if (v.y > m) { m = v.y; am = q * 4 + 1; }
            if (v.z > m) { m = v.z; am = q * 4 + 2; }
            if (v.w > m) { m = v.w; am = q * 4 + 3; }
        }
    } else {
        const float* lg = logits + (size_t)gid * C;
        __builtin_prefetch(lg + (size_t)64 * C, 0, 1);  // global_prefetch_b8
        m = lg[0]; am = 0;
        for (int c = 1; c < C; ++c) {
            float v = lg[c];
            if (v > m) { m = v; am = c; }
        }
    }
    float s = 1.0f / (1.0f + __expf(-m));   // sigmoid monotone: max/argmax agree
    scoreBits[gid] = __float_as_uint(s);    // s > 0 -> bit order == float order
    classIds[gid]  = am;
    float4 d = ((const float4*)deltas)[gid];
    if (!(d.x == -1.0f && d.y == -1.0f && d.z == -1.0f && d.w == -1.0f))
        st->allBg = 0;
}

// ---------------------------------------------------------------- radix histogram
__global__ void histKernel(const unsigned* __restrict__ bits, int N,
                           const SelState* st, unsigned* hist,
                           unsigned maskHigh, int shift) {
    __shared__ unsigned h[256];
    if (threadIdx.x < 256) h[threadIdx.x] = 0u;
    __syncthreads();
    unsigned prefix = st->prefix;
    int stride = gridDim.x * blockDim.x;
    for (int i = blockIdx.x * blockDim.x + threadIdx.x; i < N; i += stride) {
        unsigned b = bits[i];
        if ((b & maskHigh) == prefix) atomicAdd(&h[(b >> shift) & 0xFFu], 1u);
    }
    __syncthreads();
    if (threadIdx.x < 256 && h[threadIdx.x]) atomicAdd(&hist[threadIdx.x], h[threadIdx.x]);
}

// ---------------------------------------------------------------- radix select step
__global__ void selectKernel(const unsigned* hist, SelState* st, int shift) {
    if (threadIdx.x != 0 || blockIdx.x != 0) return;
    int kRem = st->kRem;
    unsigned acc = 0u;
    for (int bin = 255; bin >= 0; --bin) {
        unsigned c = hist[bin];
        if (acc + c >= (unsigned)kRem) {
            st->prefix |= ((unsigned)bin) << shift;
            st->kRem = kRem - (int)acc;
            return;
        }
        acc += c;
    }
}

// ---------------------------------------------------------------- compaction
__global__ void compactKernel(const unsigned* __restrict__ bits, int N,
                              SelState* st, unsigned long long* cand) {
    int i = blockIdx.x * blockDim.x + threadIdx.x;
    if (i >= N) return;
    unsigned b = bits[i];
    if (b >= st->prefix) {   // strictly-greater plus all ties at the threshold
        unsigned pos = atomicAdd(&st->cnt, 1u);
        if (pos < CAND_CAP)
            cand[pos] = (((unsigned long long)(~b)) << 32) | (unsigned long long)(unsigned)i;
    }
}

// ------------------------------------- bitonic sort 4096 keys (score desc, idx asc)
__global__ void sortKernel(unsigned long long* cand, const SelState* st) {
    __shared__ unsigned long long lds[CAND_CAP];
    unsigned n = st->cnt; if (n > CAND_CAP) n = CAND_CAP;
    for (int e = threadIdx.x; e < CAND_CAP; e += blockDim.x)
        lds[e] = (e < (int)n) ? cand[e] : ~0ULL;
    __syncthreads();
    for (int k = 2; k <= CAND_CAP; k <<= 1) {
        for (int j = k >> 1; j > 0; j >>= 1) {
            for (int e = threadIdx.x; e < CAND_CAP; e += blockDim.x) {
                int x = e ^ j;
                if (x > e) {
                    bool up = (e & k) == 0;
                    unsigned long long a = lds[e], b = lds[x];
                    if ((a > b) == up) { lds[e] = b; lds[x] = a; }
                }
            }
            __syncthreads();
        }
    }
    for (int e = threadIdx.x; e < CAND_CAP; e += blockDim.x) cand[e] = lds[e];
}

// ---------------------------------------------------------------- gather + decode
__global__ void gatherKernel(const unsigned long long* __restrict__ cand,
                             const float* __restrict__ deltas,
                             const float* __restrict__ locations,
                             const int* __restrict__ classIds,
                             const int* __restrict__ stridePtr,
                             const SelState* st,
                             float* __restrict__ out,
                             float* __restrict__ candBoxes,
                             int* __restrict__ candCls) {
    int k = blockIdx.x * blockDim.x + threadIdx.x;
    if (k >= NPAD) return;
    if (k >= K_PRE) {   // deterministic padding for the NMS tiles
        candBoxes[k * 4 + 0] = 0.0f; candBoxes[k * 4 + 1] = 0.0f;
        candBoxes[k * 4 + 2] = 0.0f; candBoxes[k * 4 + 3] = 0.0f;
        candCls[k] = 0;
        return;
    }
    unsigned long long key = cand[k];
    unsigned idx = (unsigned)(key & 0xFFFFFFFFull);
    float score = __uint_as_float(~(unsigned)(key >> 32));
    float sf = (float)(*stridePtr);
    float4 d = ((const float4*)deltas)[idx];
    float x = locations[(size_t)idx * 2 + 0];
    float y = locations[(size_t)idx * 2 + 1];
    float b0 = x - d.x * sf, b1 = y - d.y * sf;
    float b2 = x + d.z * sf, b3 = y + d.w * sf;
    if (st->allBg) { b0 = x; b1 = y; b2 = x; b3 = y; }
    int cls = classIds[idx];
    out[k * 4 + 0] = b0; out[k * 4 + 1] = b1;
    out[k * 4 + 2] = b2; out[k * 4 + 3] = b3;
    out[4 * K_PRE + k] = score;
    out[5 * K_PRE + k] = (float)cls;
    candBoxes[k * 4 + 0] = b0; candBoxes[k * 4 + 1] = b1;
    candBoxes[k * 4 + 2] = b2; candBoxes[k * 4 + 3] = b3;
    candCls[k] = cls;
}

// ---------------------------------------------------------------- max coordinate
__global__ void maxCoordKernel(const float* __restrict__ candBoxes, float* maxCoord) {
    __shared__ float red[1024];
    float m = -3.402823466e38f;
    for (int i = threadIdx.x; i < K_PRE * 4; i += blockDim.x) m = fmaxf(m, candBoxes[i]);
    red[threadIdx.x] = m;
    __syncthreads();
    for (int s = 512; s > 0; s >>= 1) {
        if ((int)threadIdx.x < s) red[threadIdx.x] = fmaxf(red[threadIdx.x], red[threadIdx.x + s]);
        __syncthreads();
    }
    if (threadIdx.x == 0) *maxCoord = red[0];
}

// -------------------------------------------- per-box NMS quantities (quirky coords)
__global__ void nmsPrepKernel(const float* __restrict__ candBoxes,
                              const int* __restrict__ candCls,
                              const float* __restrict__ maxCoord,
                              float* __restrict__ x1, float* __restrict__ y1,
                              float* __restrict__ x2, float* __restrict__ y2,
                              float* __restrict__ area) {
    int k = blockIdx.x * blockDim.x + threadIdx.x;
    if (k >= NPAD) return;
    if (k >= K_PRE) { x1[k] = 0; y1[k] = 0; x2[k] = 0; y2[k] = 0; area[k] = 0; return; }
    float off = (float)candCls[k] * (*maxCoord + 1.0f);
    float b0 = candBoxes[k * 4 + 0] + off, b1 = candBoxes[k * 4 + 1] + off;
    float b2 = candBoxes[k * 4 + 2] + off, b3 = candBoxes[k * 4 + 3] + off;
    x1[k] = b0; y1[k] = b3; x2[k] = b2; y2[k] = b1;   // reference's swapped-y convention
    area[k] = (b2 - b0) * (b1 - b3);
}

// -------- pairwise suppression tiles: area_i + area_j via V_WMMA_F32_16X16X4_F32
__global__ void pairKernel(const float* __restrict__ x1, const float* __restrict__ y1,
                           const float* __restrict__ x2, const float* __restrict__ y2,
                           const float* __restrict__ area, unsigned* __restrict__ S) {
    int lane = threadIdx.x;                 // 0..31, one wave32
    int i0 = blockIdx.y * 16, j0 = blockIdx.x * 16;
    bool hiHalf = lane >= 16;
    int l = lane & 15;

    v2f a, b;
    // A-matrix 16x4 f32: lanes 0-15 hold K={0,1} in v0,v1; lanes 16-31 hold K={2,3}
    a.x = hiHalf ? 0.0f : area[i0 + l];     // K=0 column: area_i
    a.y = hiHalf ? 0.0f : 1.0f;             // K=1 column: ones
    b.x = hiHalf ? 0.0f : 1.0f;             // K=0 row: ones
    b.y = hiHalf ? 0.0f : area[j0 + l];     // K=1 row: area_j
    v8f c = {};
    c = __builtin_amdgcn_wmma_f32_16x16x4_f32(
            false, a, false, b, (short)0, c, false, false);

    int mbase = hiHalf ? 8 : 0;
    int j = j0 + l;
    float xj1 = x1[j], yj1 = y1[j], xj2 = x2[j], yj2 = y2[j];
    int wshift = (j0 & 16) ? 16 : 0;
    int w = j0 >> 5;
    for (int r = 0; r < 8; ++r) {
        int i = i0 + mbase + r;
        float xi1 = x1[i], yi1 = y1[i], xi2 = x2[i], yi2 = y2[i];
        float xx1 = fmaxf(xi1, xj1), yy1 = fminf(yi1, yj1);
        float xx2 = fminf(xi2, xj2), yy2 = fmaxf(yi2, yj2);
        float inter = fabsf(xx2 - xx1) * fabsf(yy2 - yy1);
        float sumA = c[r];                          // area_i + area_j from WMMA
        float iou = inter / (sumA - inter);
        bool pred = (iou > IOU_THR) && (j > i) && (i < K_PRE) && (j < K_PRE);
        unsigned bal = __builtin_amdgcn_ballot_w32(pred);
        if (lane == 0) {
            unsigned lo   = bal & 0xFFFFu;          // row M=r   (lanes 0-15)
            unsigned hi16 = (bal >> 16) & 0xFFFFu;  // row M=r+8 (lanes 16-31)
            if (lo)   atomicOr(&S[(size_t)(i0 + r)     * SWORDS + w], lo   << wshift);
            if (hi16) atomicOr(&S[(size_t)(i0 + 8 + r) * SWORDS + w], hi16 << wshift);
        }
    }
}

// -------------------------------------- sequential NMS scan over precomputed S
__global__ void nmsScanKernel(const unsigned* __restrict__ S, float* __restrict__ keepOut) {
    __shared__ unsigned active[SWORDS];
    __shared__ int kept;
    for (int t = threadIdx.x; t < SWORDS; t += blockDim.x) active[t] = 0xFFFFFFFFu;
    __syncthreads();
    for (int i = 0; i < K_PRE; ++i) {
        if (threadIdx.x == 0) {
            int isKept = (active[i >> 5] >> (i & 31)) & 1;
            kept = isKept;
            keepOut[i] = isKept ? 1.0f : 0.0f;
        }
        __syncthreads();
        if (kept) {
            for (int t = threadIdx.x; t < SWORDS; t += blockDim.x)
                active[t] &= ~S[(size_t)i * SWORDS + t];
        }
        __syncthreads();
    }
}

// ================================================================ launch
extern "C" void kernel_launch(void* const* d_in, const int* in_sizes, int n_in,
                              void* d_out, int out_size, void* d_ws, size_t ws_size,
                              hipStream_t stream) {
    const float* deltas    = (const float*)d_in[0];
    const float* locations = (const float*)d_in[1];
    const float* logits    = (const float*)d_in[2];
    const int*   stridePtr = (const int*)d_in[3];
    float* out = (float*)d_out;

    int N = in_sizes[0] / 4;
    int C = in_sizes[2] / N;

    // ---- workspace layout ----
    char* ws = (char*)d_ws;
    size_t o = 0;
    unsigned* scoreBits = (unsigned*)(ws + o);           o += (size_t)N * 4;
    int*      classIds  = (int*)(ws + o);                o += (size_t)N * 4;
    unsigned* hist      = (unsigned*)(ws + o);           o += 4 * 256 * 4;
    SelState* st        = (SelState*)(ws + o);           o += 256;
    unsigned long long* cand = (unsigned long long*)(ws + o); o += (size_t)CAND_CAP * 8;
    float* candBoxes    = (float*)(ws + o);              o += (size_t)NPAD * 4 * 4;
    int*   candCls      = (int*)(ws + o);                o += (size_t)NPAD * 4;
    float* maxCoord     = (float*)(ws + o);              o += 256;
    float* nx1  = (float*)(ws + o);                      o += (size_t)NPAD * 4;
    float* ny1  = (float*)(ws + o);                      o += (size_t)NPAD * 4;
    float* nx2  = (float*)(ws + o);                      o += (size_t)NPAD * 4;
    float* ny2  = (float*)(ws + o);                      o += (size_t)NPAD * 4;
    float* narea = (float*)(ws + o);                     o += (size_t)NPAD * 4;
    unsigned* S = (unsigned*)(ws + o);                   o += (size_t)NPAD * SWORDS * 4;
    (void)ws_size; (void)n_in; (void)out_size;

    int blocksN = (N + TILE_R - 1) / TILE_R;

    initKernel<<<512, 256, 0, stream>>>(hist, st, S, NPAD * SWORDS);
    scoreKernel<<<blocksN, TILE_R, 0, stream>>>(logits, deltas, N, C, scoreBits, classIds, st);

    const unsigned maskHigh[4] = {0x00000000u, 0xFF000000u, 0xFFFF0000u, 0xFFFFFF00u};
    const int      shifts[4]   = {24, 16, 8, 0};
    for (int p = 0; p < 4; ++p) {
        histKernel<<<1024, 256, 0, stream>>>(scoreBits, N, st, hist + p * 256,
                                             maskHigh[p], shifts[p]);
        selectKernel<<<1, 32, 0, stream>>>(hist + p * 256, st, shifts[p]);
    }

    compactKernel<<<(N + 255) / 256, 256, 0, stream>>>(scoreBits, N, st, cand);
    sortKernel<<<1, 1024, 0, stream>>>(cand, st);
    gatherKernel<<<(NPAD + 255) / 256, 256, 0, stream>>>(cand, deltas, locations, classIds,
                                                         stridePtr, st, out, candBoxes, candCls);
    maxCoordKernel<<<1, 1024, 0, stream>>>(candBoxes, maxCoord);
    nmsPrepKernel<<<(NPAD + 255) / 256, 256, 0, stream>>>(candBoxes, candCls, maxCoord,
                                                          nx1, ny1, nx2, ny2, narea);
    pairKernel<<<dim3(NPAD / 16, NPAD / 16), 32, 0, stream>>>(nx1, ny1, nx2, ny2, narea, S);
    nmsScanKernel<<<1, 64, 0, stream>>>(S, out + 6 * K_PRE);
}